// DMoNModel_66039417143756
// MI455X (gfx1250) — compile-verified
//
#include <hip/hip_runtime.h>
#include <math.h>

typedef float v2f __attribute__((ext_vector_type(2)));
typedef float v8f __attribute__((ext_vector_type(8)));

#define IN_DIM 128
#define HID    128
#define KOUT   16

// ---------------------------------------------------------------------------
// init: deg = 1.0 (self loop), AGG1 = 0, AGG2 = 0
// ---------------------------------------------------------------------------
__global__ void init_kernel(float* __restrict__ deg, float* __restrict__ agg1,
                            float* __restrict__ agg2, int N) {
    int i = blockIdx.x * blockDim.x + threadIdx.x;
    if (i < N * HID) agg1[i] = 0.0f;
    if (i < N * KOUT) agg2[i] = 0.0f;
    if (i < N) deg[i] = 1.0f;
}

// ---------------------------------------------------------------------------
// degree accumulation over edge dst
// ---------------------------------------------------------------------------
__global__ void deg_kernel(const int* __restrict__ dst, float* __restrict__ deg, int E) {
    int e = blockIdx.x * blockDim.x + threadIdx.x;
    if (e < E) atomicAdd(&deg[dst[e]], 1.0f);
}

// deg -> deg^{-1/2} in place (deg >= 1 always due to self loops)
__global__ void rsqrt_kernel(float* __restrict__ deg, int N) {
    int i = blockIdx.x * blockDim.x + threadIdx.x;
    if (i < N) deg[i] = rsqrtf(deg[i]);
}

// ---------------------------------------------------------------------------
// H = X(50000x128) @ W1(128x128), fp32 WMMA 16x16x4.
// grid.x = N/16 (M tiles), 8 waves/block = 8 N tiles (HID = 8*16).
// ---------------------------------------------------------------------------
__global__ void gemm1_kernel(const float* __restrict__ X, const float* __restrict__ W,
                             float* __restrict__ H) {
    const int wave = threadIdx.x >> 5;
    const int lane = threadIdx.x & 31;
    const int hi   = lane >> 4;     // lane half: selects K pair / M half
    const int r    = lane & 15;
    const int m0   = blockIdx.x * 16;
    const int n0   = wave * 16;

    v8f c = {};
    const float* xrow = X + (size_t)(m0 + r) * IN_DIM;
    for (int k = 0; k < IN_DIM; k += 4) {
        v2f a, b;
        a.x = xrow[k + 2 * hi];
        a.y = xrow[k + 2 * hi + 1];
        b.x = W[(size_t)(k + 2 * hi) * HID + n0 + r];
        b.y = W[(size_t)(k + 2 * hi + 1) * HID + n0 + r];
        c = __builtin_amdgcn_wmma_f32_16x16x4_f32(false, a, false, b,
                                                  (short)0, c, false, false);
    }
    // C/D layout: VGPR i -> rows m0+i (lanes 0-15) / m0+8+i (lanes 16-31), col n0+r
    float* hbase = H + (size_t)m0 * HID + n0;
#pragma unroll
    for (int i = 0; i < 8; ++i)
        hbase[(size_t)(i + 8 * hi) * HID + r] = c[i];
}

// ---------------------------------------------------------------------------
// H2 = A(50000x128) @ W2(128x16), fp32 WMMA. One wave per 16-row tile.
// ---------------------------------------------------------------------------
__global__ void gemm2_kernel(const float* __restrict__ A, const float* __restrict__ W,
                             float* __restrict__ H2, int mtiles) {
    const int wave = threadIdx.x >> 5;
    const int mt   = blockIdx.x * 8 + wave;
    if (mt >= mtiles) return;
    const int lane = threadIdx.x & 31;
    const int hi   = lane >> 4;
    const int r    = lane & 15;
    const int m0   = mt * 16;

    v8f c = {};
    const float* arow = A + (size_t)(m0 + r) * HID;
    for (int k = 0; k < HID; k += 4) {
        v2f a, b;
        a.x = arow[k + 2 * hi];
        a.y = arow[k + 2 * hi + 1];
        b.x = W[(size_t)(k + 2 * hi) * KOUT + r];
        b.y = W[(size_t)(k + 2 * hi + 1) * KOUT + r];
        c = __builtin_amdgcn_wmma_f32_16x16x4_f32(false, a, false, b,
                                                  (short)0, c, false, false);
    }
    float* obase = H2 + (size_t)m0 * KOUT;
#pragma unroll
    for (int i = 0; i < 8; ++i)
        obase[(size_t)(i + 8 * hi) * KOUT + r] = c[i];
}

// ---------------------------------------------------------------------------
// Layer-1 edge scatter: one wave per edge, lane covers dims l, l+32, l+64, l+96
// ---------------------------------------------------------------------------
__global__ void scatter1_kernel(const int* __restrict__ src, const int* __restrict__ dst,
                                const float* __restrict__ H, const float* __restrict__ dinv,
                                float* __restrict__ agg1, int E) {
    int tid  = blockIdx.x * blockDim.x + threadIdx.x;
    int e    = tid >> 5;
    int lane = tid & 31;
    if (e >= E) return;
    int s = src[e], d = dst[e];
    float nrm = dinv[s] * dinv[d];
    const float* hrow = H + (size_t)s * HID;
    float* orow = agg1 + (size_t)d * HID;
#pragma unroll
    for (int i = 0; i < 4; ++i) {
        int dim = lane + 32 * i;
        atomicAdd(&orow[dim], hrow[dim] * nrm);
    }
}

// fold self-loop + bias, then ReLU:  agg1 = relu(agg1 + H*dinv^2 + b1)
__global__ void relu_kernel(float* __restrict__ agg1, const float* __restrict__ H,
                            const float* __restrict__ dinv, const float* __restrict__ b1,
                            int N) {
    int idx = blockIdx.x * blockDim.x + threadIdx.x;
    if (idx >= N * HID) return;
    int i = idx >> 7;           // / HID
    int d = idx & (HID - 1);
    float nr = dinv[i] * dinv[i];
    float v = agg1[idx] + H[idx] * nr + b1[d];
    agg1[idx] = fmaxf(v, 0.0f);
}

// ---------------------------------------------------------------------------
// Layer-2 edge scatter: one thread per (edge, dim); 16 dims per edge
// ---------------------------------------------------------------------------
__global__ void scatter2_kernel(const int* __restrict__ src, const int* __restrict__ dst,
                                const float* __restrict__ H2, const float* __restrict__ dinv,
                                float* __restrict__ agg2, int E) {
    int tid = blockIdx.x * blockDim.x + threadIdx.x;
    int e = tid >> 4;
    int d = tid & 15;
    if (e >= E) return;
    int s = src[e], t = dst[e];
    float nrm = dinv[s] * dinv[t];
    atomicAdd(&agg2[(size_t)t * KOUT + d], H2[(size_t)s * KOUT + d] * nrm);
}

// self-loop + bias + softmax -> out
__global__ void softmax_kernel(const float* __restrict__ agg2, const float* __restrict__ H2,
                               const float* __restrict__ dinv, const float* __restrict__ b2,
                               float* __restrict__ out, int N) {
    int i = blockIdx.x * blockDim.x + threadIdx.x;
    if (i >= N) return;
    float nr = dinv[i] * dinv[i];
    float v[KOUT];
    float mx = -3.0e38f;
    const size_t base = (size_t)i * KOUT;
#pragma unroll
    for (int d = 0; d < KOUT; ++d) {
        v[d] = agg2[base + d] + H2[base + d] * nr + b2[d];
        mx = fmaxf(mx, v[d]);
    }
    float sum = 0.0f;
#pragma unroll
    for (int d = 0; d < KOUT; ++d) {
        v[d] = __expf(v[d] - mx);
        sum += v[d];
    }
    float inv = 1.0f / sum;
#pragma unroll
    for (int d = 0; d < KOUT; ++d)
        out[base + d] = v[d] * inv;
}

// ---------------------------------------------------------------------------
extern "C" void kernel_launch(void* const* d_in, const int* in_sizes, int n_in,
                              void* d_out, int out_size, void* d_ws, size_t ws_size,
                              hipStream_t stream) {
    const float* X  = (const float*)d_in[0];
    const int*   EI = (const int*)d_in[1];   // [2, E] row-major (jax default x64-off -> int32)
    const float* W1 = (const float*)d_in[2];
    const float* B1 = (const float*)d_in[3];
    const float* W2 = (const float*)d_in[4];
    const float* B2 = (const float*)d_in[5];

    const int N = in_sizes[0] / IN_DIM;
    const int E = in_sizes[1] / 2;
    const int* srcI = EI;
    const int* dstI = EI + E;

    // workspace layout (floats)
    float* ws   = (float*)d_ws;
    float* dinv = ws;                          // N      (deg, then deg^-1/2)
    float* H    = dinv + N;                    // N*HID
    float* AGG1 = H + (size_t)N * HID;         // N*HID
    float* H2   = AGG1 + (size_t)N * HID;      // N*KOUT
    float* AGG2 = H2 + (size_t)N * KOUT;       // N*KOUT
    float* OUT  = (float*)d_out;

    const int B = 256;

    init_kernel<<<(N * HID + B - 1) / B, B, 0, stream>>>(dinv, AGG1, AGG2, N);
    deg_kernel<<<(E + B - 1) / B, B, 0, stream>>>(dstI, dinv, E);
    rsqrt_kernel<<<(N + B - 1) / B, B, 0, stream>>>(dinv, N);

    // H = X @ W1  (N/16 M-tiles, 8 waves = 8 N-tiles per block)
    gemm1_kernel<<<N / 16, B, 0, stream>>>(X, W1, H);

    // layer 1 scatter: one wave per edge -> E*32 threads
    scatter1_kernel<<<(E * 8 + (B / 32 - 1)) / (B / 32) , B, 0, stream>>>(
        srcI, dstI, H, dinv, AGG1, E);
    // note: grid = (E*8 + 7) / 8 = E blocks of 256 threads; waves with e >= E exit
    relu_kernel<<<(N * HID + B - 1) / B, B, 0, stream>>>(AGG1, H, dinv, B1, N);

    // H2 = relu(AGG1) @ W2
    const int mtiles = N / 16;
    gemm2_kernel<<<(mtiles + 7) / 8, B, 0, stream>>>(AGG1, W2, H2, mtiles);

    // layer 2 scatter: one thread per (edge, dim)
    long long t2 = (long long)E * KOUT;
    scatter2_kernel<<<(int)((t2 + B - 1) / B), B, 0, stream>>>(srcI, dstI, H2, dinv, AGG2, E);

    softmax_kernel<<<(N + B - 1) / B, B, 0, stream>>>(AGG2, H2, dinv, B2, OUT, N);
}